// EfficientEdgeConv_36481452213111
// MI455X (gfx1250) — compile-verified
//
#include <hip/hip_runtime.h>
#include <float.h>

typedef __attribute__((ext_vector_type(2))) float v2f;
typedef __attribute__((ext_vector_type(8))) float v8f;

#define B_     4
#define C_     64
#define N_     8192
#define K_     20
#define OUT_   64
#define EPS_   1e-5f
#define SLOPE_ 0.2f
#define KNN_WAVES_ 4          // waves per kNN block; block owns 128 rows

// ---------- xx[b,n] = sum_c x[b,c,n]^2 ----------
__global__ __launch_bounds__(256) void xx_kernel(const float* __restrict__ x,
                                                 float* __restrict__ xx) {
  int i = blockIdx.x * 256 + threadIdx.x;       // i = b*N + n
  int b = i >> 13, n = i & (N_ - 1);
  const float* p = x + (size_t)b * C_ * N_ + n;
  float s = 0.f;
#pragma unroll
  for (int c = 0; c < C_; ++c) { float v = p[(size_t)c * N_]; s += v * v; }
  xx[i] = s;
}

// ---------- fused GEMM (x^T x) + per-row top-K=20 ----------
// 4 waves / block; each wave owns 32 rows (two 16x16 f32 WMMA accumulators).
// Double-buffered LDS B panel: tile t+1 is fetched from global into registers
// while tile t's WMMAs run; all 16 B fragments are preloaded into VGPRs so
// the DS pipeline stays 16 deep instead of stalling per step.
// Ranking key s = 2*inner[i,j] - xx[j]  (row-constant -xx[i] dropped).
__global__ __launch_bounds__(32 * KNN_WAVES_) void knn_kernel(
    const float* __restrict__ x, const float* __restrict__ xx,
    int* __restrict__ knn_idx) {
  __shared__ float bpanel[2][C_ * 16];                 // [c][n], stride 16: conflict-free
  __shared__ float sc[KNN_WAVES_ * 32 * 17];           // per-wave padded score scratch
  const int b    = blockIdx.y;
  const int tid  = threadIdx.x;
  const int wave = tid >> 5;
  const int lane = tid & 31;
  const int row0 = blockIdx.x * (32 * KNN_WAVES_) + wave * 32;
  const int lm   = lane & 15;
  const int ksel = (lane >> 4) << 1;                   // K offset 0/2 (A/B frag layout)
  const int hsel = (lane >> 4) << 3;                   // row offset 0/8 (C/D layout)
  const float* xb = x + (size_t)b * C_ * N_;
  float* mysc = &sc[wave * 32 * 17];

  // cooperative-staging coordinates: 2 float4 per thread (1024 floats total)
  const int c0s = (tid + 0)   >> 2, n0s = ((tid + 0)   & 3) << 2;
  const int c1s = (tid + 128) >> 2, n1s = ((tid + 128) & 3) << 2;

  // A fragments for this wave's two 16-row tiles (loop-invariant)
  v2f A0[16], A1[16];
#pragma unroll
  for (int s = 0; s < 16; ++s) {
    int c = 4 * s + ksel;
    A0[s].x = xb[(size_t)c * N_ + row0 + lm];
    A0[s].y = xb[(size_t)(c + 1) * N_ + row0 + lm];
    A1[s].x = xb[(size_t)c * N_ + row0 + 16 + lm];
    A1[s].y = xb[(size_t)(c + 1) * N_ + row0 + 16 + lm];
  }

  // Register-resident top-K (fully unrolled static indexing -> stays in VGPRs)
  float tv[K_]; int ti[K_];
#pragma unroll
  for (int q = 0; q < K_; ++q) { tv[q] = -FLT_MAX; ti[q] = 0; }
  float tmin = -FLT_MAX; int tminpos = 0;

  // prologue: stage tile 0 into panel 0
  {
    float4 s0 = *(const float4*)(xb + (size_t)c0s * N_ + n0s);
    float4 s1 = *(const float4*)(xb + (size_t)c1s * N_ + n1s);
    *(float4*)(&bpanel[0][c0s * 16 + n0s]) = s0;
    *(float4*)(&bpanel[0][c1s * 16 + n1s]) = s1;
  }
  __syncthreads();

  for (int j0 = 0; j0 < N_; j0 += 16) {
    const int buf = (j0 >> 4) & 1;
    const float* cur = bpanel[buf];
    const bool more = (j0 + 16) < N_;

    // prefetch next tile from global into registers (overlaps WMMA below)
    float4 s0 = {}, s1 = {};
    if (more) {
      s0 = *(const float4*)(xb + (size_t)c0s * N_ + (j0 + 16) + n0s);
      s1 = *(const float4*)(xb + (size_t)c1s * N_ + (j0 + 16) + n1s);
    }

    // preload all 16 B fragments -> 16-deep DS pipeline
    v2f Bf[16];
#pragma unroll
    for (int s = 0; s < 16; ++s) {
      int c = 4 * s + ksel;
      Bf[s].x = cur[c * 16 + lm];
      Bf[s].y = cur[(c + 1) * 16 + lm];
    }

    v8f acc0 = {}; v8f acc1 = {};
#pragma unroll
    for (int s = 0; s < 16; ++s) {
      acc0 = __builtin_amdgcn_wmma_f32_16x16x4_f32(false, A0[s], false, Bf[s],
                                                   (short)0, acc0, false, false);
      acc1 = __builtin_amdgcn_wmma_f32_16x16x4_f32(false, A1[s], false, Bf[s],
                                                   (short)0, acc1, false, false);
    }

    // publish prefetched tile into the other panel
    if (more) {
      *(float4*)(&bpanel[1 - buf][c0s * 16 + n0s]) = s0;
      *(float4*)(&bpanel[1 - buf][c1s * 16 + n1s]) = s1;
    }

    float xxj = xx[b * N_ + j0 + lm];
#pragma unroll
    for (int r = 0; r < 8; ++r) {
      mysc[(r + hsel) * 17 + lm]      = 2.f * acc0[r] - xxj;   // rows 0..15
      mysc[(16 + r + hsel) * 17 + lm] = 2.f * acc1[r] - xxj;   // rows 16..31
    }
    // intra-wave LDS store->load: hardware keeps DS ops of a wave in order
    const float* myrow = &mysc[lane * 17];             // lane owns row row0+lane
#pragma unroll 1
    for (int cc = 0; cc < 16; ++cc) {
      float sv = myrow[cc];
      if (sv > tmin) {
        int jg = j0 + cc;
#pragma unroll
        for (int q = 0; q < K_; ++q)
          if (q == tminpos) { tv[q] = sv; ti[q] = jg; }
        tmin = tv[0]; tminpos = 0;
#pragma unroll
        for (int q = 1; q < K_; ++q)
          if (tv[q] < tmin) { tmin = tv[q]; tminpos = q; }
      }
    }
    // one barrier per tile: publishes next panel AND guarantees all waves'
    // fragment reads of 'cur' finished before it is overwritten next round
    __syncthreads();
  }

  int* op = knn_idx + ((size_t)b * N_ + row0 + lane) * K_;
#pragma unroll
  for (int q = 0; q < K_; ++q) op[q] = ti[q];
}

// ---------- y1 = W1 @ x ; y2 = (W2 - W1) @ x  (per batch, WMMA) ----------
__global__ __launch_bounds__(32) void proj_kernel(const float* __restrict__ x,
                                                  const float* __restrict__ W,
                                                  float* __restrict__ y1,
                                                  float* __restrict__ y2) {
  const int b  = blockIdx.z;
  const int ot = blockIdx.y;                    // output-channel tile (0..3)
  const int n0 = blockIdx.x * 16;
  const int lane = threadIdx.x;
  const int lm   = lane & 15;
  const int ksel = (lane >> 4) << 1;
  const int hsel = (lane >> 4) << 3;
  const float* xb = x + (size_t)b * C_ * N_;
  const float* Wr = W + (size_t)(ot * 16 + lm) * (2 * C_);

  v2f Aw[16], Ad[16];
#pragma unroll
  for (int s = 0; s < 16; ++s) {
    int c = 4 * s + ksel;
    float w1x = Wr[c], w1y = Wr[c + 1];
    Aw[s].x = w1x;                 Aw[s].y = w1y;
    Ad[s].x = Wr[C_ + c] - w1x;    Ad[s].y = Wr[C_ + c + 1] - w1y;
  }
  v8f a1 = {}; v8f a2 = {};
#pragma unroll
  for (int s = 0; s < 16; ++s) {
    int c = 4 * s + ksel;
    v2f Bf;
    Bf.x = xb[(size_t)c * N_ + n0 + lm];
    Bf.y = xb[(size_t)(c + 1) * N_ + n0 + lm];
    a1 = __builtin_amdgcn_wmma_f32_16x16x4_f32(false, Aw[s], false, Bf,
                                               (short)0, a1, false, false);
    a2 = __builtin_amdgcn_wmma_f32_16x16x4_f32(false, Ad[s], false, Bf,
                                               (short)0, a2, false, false);
  }
#pragma unroll
  for (int r = 0; r < 8; ++r) {
    int o = ot * 16 + r + hsel;
    size_t off = ((size_t)b * OUT_ + o) * N_ + n0 + lm;
    y1[off] = a1[r];
    y2[off] = a2[r];
  }
}

// ---------- gather-add; 8 channels per thread reuse one idx read ----------
__global__ __launch_bounds__(256) void stats_kernel(const float* __restrict__ y1,
                                                    const float* __restrict__ y2,
                                                    const int* __restrict__ idx,
                                                    float* __restrict__ mx,
                                                    float* __restrict__ mn,
                                                    float* __restrict__ stats) {
  const int b  = blockIdx.z;
  const int o0 = blockIdx.y * 8;
  const int n  = blockIdx.x * 256 + threadIdx.x;
  const int* ip = idx + ((size_t)b * N_ + n) * K_;
  int jj[K_];
#pragma unroll
  for (int k = 0; k < K_; ++k) jj[k] = ip[k];

  float sum[8], ssq[8];
#pragma unroll
  for (int oo = 0; oo < 8; ++oo) {
    const int o = o0 + oo;
    const float* y1row = y1 + ((size_t)b * OUT_ + o) * N_;
    const float s2 = y2[((size_t)b * OUT_ + o) * N_ + n];
    float vmax = -FLT_MAX, vmin = FLT_MAX, s = 0.f, q = 0.f;
#pragma unroll
    for (int k = 0; k < K_; ++k) {
      float v = y1row[jj[k]] + s2;
      vmax = fmaxf(vmax, v);
      vmin = fminf(vmin, v);
      s += v; q += v * v;
    }
    size_t off = ((size_t)b * OUT_ + o) * N_ + n;
    mx[off] = vmax; mn[off] = vmin;
    sum[oo] = s; ssq[oo] = q;
  }

  // wave32 butterfly reduction of the 16 partials, then lane0 atomics
#pragma unroll
  for (int oo = 0; oo < 8; ++oo) {
#pragma unroll
    for (int m = 16; m > 0; m >>= 1) {
      sum[oo] += __shfl_xor(sum[oo], m);
      ssq[oo] += __shfl_xor(ssq[oo], m);
    }
  }
  if ((threadIdx.x & 31) == 0) {
#pragma unroll
    for (int oo = 0; oo < 8; ++oo) {
      atomicAdd(&stats[2 * (o0 + oo)],     sum[oo]);
      atomicAdd(&stats[2 * (o0 + oo) + 1], ssq[oo]);
    }
  }
}

// ---------- finalize: BN affine + LeakyReLU applied to max (or min) ----------
__global__ __launch_bounds__(256) void out_kernel(const float* __restrict__ mx,
                                                  const float* __restrict__ mn,
                                                  const float* __restrict__ stats,
                                                  const float* __restrict__ gamma,
                                                  const float* __restrict__ beta,
                                                  float* __restrict__ out) {
  int i = blockIdx.x * 256 + threadIdx.x;       // i = ((b*OUT)+o)*N + n
  int o = (i >> 13) & (OUT_ - 1);
  const float invc = 1.f / (float)(B_ * N_ * K_);
  float mean = stats[2 * o] * invc;
  float var  = stats[2 * o + 1] * invc - mean * mean;
  float a  = gamma[o] * rsqrtf(var + EPS_);
  float bb = beta[o] - mean * a;
  float v  = (a >= 0.f) ? mx[i] : mn[i];        // monotone transform commutes with max_k
  float y  = a * v + bb;
  out[i] = (y > 0.f) ? y : SLOPE_ * y;
}

extern "C" void kernel_launch(void* const* d_in, const int* in_sizes, int n_in,
                              void* d_out, int out_size, void* d_ws, size_t ws_size,
                              hipStream_t stream) {
  const float* x     = (const float*)d_in[0];
  const float* W     = (const float*)d_in[1];
  const float* gamma = (const float*)d_in[2];
  const float* beta  = (const float*)d_in[3];
  float* out = (float*)d_out;

  char* ws = (char*)d_ws;
  float* xx_buf = (float*)(ws + 0);              //  131072 B
  int*   idx    = (int*)  (ws + 131072);         // 2621440 B
  float* y1     = (float*)(ws + 2752512);        // 8388608 B
  float* y2     = (float*)(ws + 11141120);       // 8388608 B
  float* mx     = (float*)(ws + 19529728);       // 8388608 B
  float* mn     = (float*)(ws + 27918336);       // 8388608 B
  float* stats  = (float*)(ws + 36306944);       //     512 B

  hipMemsetAsync(stats, 0, 2 * OUT_ * sizeof(float), stream);

  xx_kernel  <<<dim3(B_ * N_ / 256),                    dim3(256), 0, stream>>>(x, xx_buf);
  knn_kernel <<<dim3(N_ / (32 * KNN_WAVES_), B_),       dim3(32 * KNN_WAVES_), 0, stream>>>(x, xx_buf, idx);
  proj_kernel<<<dim3(N_ / 16, OUT_ / 16, B_),           dim3(32),  0, stream>>>(x, W, y1, y2);
  stats_kernel<<<dim3(N_ / 256, OUT_ / 8, B_),          dim3(256), 0, stream>>>(y1, y2, idx, mx, mn, stats);
  out_kernel <<<dim3(B_ * OUT_ * N_ / 256),             dim3(256), 0, stream>>>(mx, mn, stats, gamma, beta, out);
}